// SSM_68478958568092
// MI455X (gfx1250) — compile-verified
//
#include <hip/hip_runtime.h>

typedef __attribute__((ext_vector_type(2))) float v2f;
typedef __attribute__((ext_vector_type(8))) float v8f;

constexpr int D     = 512;
constexpr int T     = 4096;
constexpr int NC    = 64;   // chunks
constexpr int CLEN  = 64;   // steps per chunk (NC*CLEN == T)

// ---------- LDS swizzle for the 16x512 state buffer (bank-conflict free at stride 512)
__device__ __forceinline__ int hswz(int r, int c) { return r * D + (c ^ (r << 2)); }

// A-fragment (16x4 f32) of the state h from LDS.
// lanes 0-15: row=lane, K={k,k+1}; lanes 16-31: row=lane-16, K={k+2,k+3}
__device__ __forceinline__ v2f lds_afrag(const float* hb, int k, int lane) {
    int r = lane & 15;
    int c = k + ((lane >> 4) << 1);           // even; swizzle keeps the pair contiguous
    return *(const v2f*)(hb + r * D + (c ^ (r << 2)));
}

// B-fragment (4x16 f32) of Wᵀ where W is row-major [e][d]: Bmat[k'][n] = W[nbase+n][k+k']
__device__ __forceinline__ v2f bfragT(const float* __restrict__ W, int nbase, int k, int lane) {
    int rn = nbase + (lane & 15);
    int ck = k + ((lane >> 4) << 1);
    return *(const v2f*)(W + (size_t)rn * D + ck);
}

// One recurrence GEMM: acc[t] (16x16 tiles, N = wave*64 + t*16) += h(16x512) @ Wᵀ(512x512)
__device__ __forceinline__ void gemm_step(const float* __restrict__ W, const float* hb,
                                          int wave, int lane, v8f acc[4]) {
    int nb = wave * 64;
    for (int k = 0; k < D; k += 4) {
        v2f a = lds_afrag(hb, k, lane);
#pragma unroll
        for (int t = 0; t < 4; ++t) {
            v2f b = bfragT(W, nb + t * 16, k, lane);
            acc[t] = __builtin_amdgcn_wmma_f32_16x16x4_f32(
                false, a, false, b, (short)0, acc[t], false, false);
        }
    }
}

// Scatter a 16x16 f32 C/D fragment into the swizzled LDS state buffer.
// lanes 0-15: rows 0-7 (vgpr i -> M=i), lanes 16-31: rows 8-15.
__device__ __forceinline__ void store_h(float* hb, int lane, int nbase, v8f acc) {
    int n  = lane & 15;
    int mb = (lane >> 4) << 3;
#pragma unroll
    for (int i = 0; i < 8; ++i) {
        int r = mb + i, c = nbase + n;
        hb[r * D + (c ^ (r << 2))] = acc[i];
    }
}

// ---------- Pass 1: U[b,t,:] = x[b,t,:] @ Bwᵀ   (M=32768, N=K=512), U == d_out
__global__ __launch_bounds__(256) void k_inproj(const float* __restrict__ x,
                                                const float* __restrict__ Bw,
                                                float* __restrict__ U) {
    int wave = threadIdx.x >> 5, lane = threadIdx.x & 31;
    int task  = blockIdx.x * 8 + wave;        // 16384 wave-tasks: 16 rows x 64 cols each
    int rbase = (task >> 3) * 16;
    int nb64  = (task & 7) * 64;

    v8f acc[4] = {};
    for (int k = 0; k < D; k += 4) {
        int r = rbase + (lane & 15);
        int c = k + ((lane >> 4) << 1);
        v2f a = *(const v2f*)(x + (size_t)r * D + c);
#pragma unroll
        for (int t = 0; t < 4; ++t) {
            v2f b = bfragT(Bw, nb64 + t * 16, k, lane);
            acc[t] = __builtin_amdgcn_wmma_f32_16x16x4_f32(
                false, a, false, b, (short)0, acc[t], false, false);
        }
    }
    int n = lane & 15, mb = (lane >> 4) << 3;
#pragma unroll
    for (int t = 0; t < 4; ++t)
#pragma unroll
        for (int i = 0; i < 8; ++i)
            U[(size_t)(rbase + mb + i) * D + nb64 + t * 16 + n] = acc[t][i];
}

// ---------- Squaring GEMM: Mout = Min @ Min (512^3), standard (non-transposed) B operand
__global__ __launch_bounds__(256) void k_sq(const float* __restrict__ Min,
                                            float* __restrict__ Mout) {
    int wave = threadIdx.x >> 5, lane = threadIdx.x & 31;
    int task  = blockIdx.x * 8 + wave;        // 256 wave-tasks
    int rbase = (task >> 3) * 16;
    int nb64  = (task & 7) * 64;

    v8f acc[4] = {};
    for (int k = 0; k < D; k += 4) {
        int r = rbase + (lane & 15);
        int c = k + ((lane >> 4) << 1);
        v2f a = *(const v2f*)(Min + (size_t)r * D + c);
        int rk = c;                            // same K offsets for the B operand
#pragma unroll
        for (int t = 0; t < 4; ++t) {
            int cn = nb64 + t * 16 + (lane & 15);
            v2f b;
            b.x = Min[(size_t)rk * D + cn];
            b.y = Min[(size_t)(rk + 1) * D + cn];
            acc[t] = __builtin_amdgcn_wmma_f32_16x16x4_f32(
                false, a, false, b, (short)0, acc[t], false, false);
        }
    }
    int n = lane & 15, mb = (lane >> 4) << 3;
#pragma unroll
    for (int t = 0; t < 4; ++t)
#pragma unroll
        for (int i = 0; i < 8; ++i)
            Mout[(size_t)(rbase + mb + i) * D + nb64 + t * 16 + n] = acc[t][i];
}

// ---------- Pass 2: two chunks per WG (rows 0-7 = chunk c0, rows 8-15 = chunk c1),
//            recurrence from zero state; emit only each chunk's final state.
__global__ __launch_bounds__(256) void k_local(const float* __restrict__ A,
                                               const float* __restrict__ U,
                                               float* __restrict__ pend) {
    __shared__ float hbuf[2][16 * D];         // 64 KB double buffer
    int c0 = blockIdx.x * 2, c1 = c0 + 1;
    int tid = threadIdx.x, wave = tid >> 5, lane = tid & 31;

    for (int i = tid; i < 16 * D; i += 256) hbuf[0][i] = 0.f;
    __syncthreads();

    int cur = 0;
    int n = lane & 15;
    for (int j = 0; j < CLEN; ++j) {
        int t = ((lane < 16) ? c0 : c1) * CLEN + j;   // per-half time index
        v8f acc[4] = {};
        gemm_step(A, hbuf[cur], wave, lane, acc);
#pragma unroll
        for (int tt = 0; tt < 4; ++tt)
#pragma unroll
            for (int i = 0; i < 8; ++i)                // vgpr i -> batch i of this half's chunk
                acc[tt][i] += U[((size_t)i * T + t) * D + wave * 64 + tt * 16 + n];
#pragma unroll
        for (int tt = 0; tt < 4; ++tt)
            store_h(hbuf[cur ^ 1], lane, wave * 64 + tt * 16, acc[tt]);
        __syncthreads();
        cur ^= 1;
    }
    for (int i = tid; i < 16 * D; i += 256) {
        int r = i >> 9, c = i & (D - 1);
        int ch = (r < 8) ? c0 : c1;
        pend[((size_t)ch * 8 + (r & 7)) * D + c] = hbuf[cur][hswz(r, c)];
    }
}

// ---------- Pass 3: sequential scan over chunk states: s_{n+1} = A^64 s_n + pend_n
//            (single serial chain; rows 8-15 stay zero)
__global__ __launch_bounds__(256) void k_scan(const float* __restrict__ Ac,
                                              const float* __restrict__ h0,
                                              const float* __restrict__ pend,
                                              float* __restrict__ sst) {
    __shared__ float hbuf[2][16 * D];
    int tid = threadIdx.x, wave = tid >> 5, lane = tid & 31;

    for (int i = tid; i < 16 * D; i += 256) {
        int r = i >> 9, c = i & (D - 1);
        hbuf[0][hswz(r, c)] = (r < 8) ? h0[(size_t)r * D + c] : 0.f;
    }
    __syncthreads();

    int cur = 0;
    for (int nck = 0; nck < NC; ++nck) {
        // emit s_nck = state entering chunk nck (real rows 0-7 only)
        for (int i = tid; i < 8 * D; i += 256) {
            int r = i >> 9, c = i & (D - 1);
            sst[(size_t)nck * 8 * D + i] = hbuf[cur][hswz(r, c)];
        }
        v8f acc[4] = {};
        gemm_step(Ac, hbuf[cur], wave, lane, acc);
        int n = lane & 15;
        if (lane < 16) {                      // pend has 8 rows per chunk
#pragma unroll
            for (int tt = 0; tt < 4; ++tt)
#pragma unroll
                for (int i = 0; i < 8; ++i)
                    acc[tt][i] += pend[((size_t)nck * 8 + i) * D + wave * 64 + tt * 16 + n];
        }
#pragma unroll
        for (int tt = 0; tt < 4; ++tt)
            store_h(hbuf[cur ^ 1], lane, wave * 64 + tt * 16, acc[tt]);
        __syncthreads();
        cur ^= 1;
    }
}

// ---------- Pass 4: two chunks per WG, recurrence from exact start states,
//            writing the trajectory h_t over U in place.
__global__ __launch_bounds__(256) void k_final(const float* __restrict__ A,
                                               const float* __restrict__ sst,
                                               float* __restrict__ U) {
    __shared__ float hbuf[2][16 * D];
    int c0 = blockIdx.x * 2, c1 = c0 + 1;
    int tid = threadIdx.x, wave = tid >> 5, lane = tid & 31;

    for (int i = tid; i < 16 * D; i += 256) {
        int r = i >> 9, c = i & (D - 1);
        int ch = (r < 8) ? c0 : c1;
        hbuf[0][hswz(r, c)] = sst[((size_t)ch * 8 + (r & 7)) * D + c];
    }
    __syncthreads();

    int cur = 0;
    int n = lane & 15;
    for (int j = 0; j < CLEN; ++j) {
        int t = ((lane < 16) ? c0 : c1) * CLEN + j;
        v8f acc[4] = {};
        gemm_step(A, hbuf[cur], wave, lane, acc);
#pragma unroll
        for (int tt = 0; tt < 4; ++tt)
#pragma unroll
            for (int i = 0; i < 8; ++i)
                acc[tt][i] += U[((size_t)i * T + t) * D + wave * 64 + tt * 16 + n];
#pragma unroll
        for (int tt = 0; tt < 4; ++tt)
            store_h(hbuf[cur ^ 1], lane, wave * 64 + tt * 16, acc[tt]);
#pragma unroll
        for (int tt = 0; tt < 4; ++tt)         // h_t out (overwrites consumed U_t)
#pragma unroll
            for (int i = 0; i < 8; ++i)
                U[((size_t)i * T + t) * D + wave * 64 + tt * 16 + n] = acc[tt][i];
        __syncthreads();
        cur ^= 1;
    }
}

extern "C" void kernel_launch(void* const* d_in, const int* in_sizes, int n_in,
                              void* d_out, int out_size, void* d_ws, size_t ws_size,
                              hipStream_t stream) {
    (void)in_sizes; (void)n_in; (void)out_size; (void)ws_size;
    const float* x  = (const float*)d_in[0];   // [8,4096,512]
    const float* h0 = (const float*)d_in[1];   // [8,512]
    const float* A  = (const float*)d_in[2];   // [512,512]
    const float* Bw = (const float*)d_in[3];   // [512,512]
    float* U  = (float*)d_out;                 // [8,4096,512], holds Bx then h
    float* ws = (float*)d_ws;

    float* ping = ws;                          // 512*512
    float* pong = ws + D * D;                  // 512*512
    float* pend = ws + 2 * D * D;              // NC*8*512
    float* sst  = pend + NC * 8 * D;           // NC*8*512    (4 MiB total)

    k_inproj<<<2048, 256, 0, stream>>>(x, Bw, U);
    k_sq<<<32, 256, 0, stream>>>(A, ping);     // A^2
    k_sq<<<32, 256, 0, stream>>>(ping, pong);  // A^4
    k_sq<<<32, 256, 0, stream>>>(pong, ping);  // A^8
    k_sq<<<32, 256, 0, stream>>>(ping, pong);  // A^16
    k_sq<<<32, 256, 0, stream>>>(pong, ping);  // A^32
    k_sq<<<32, 256, 0, stream>>>(ping, pong);  // A^64
    k_local<<<NC / 2, 256, 0, stream>>>(A, U, pend);
    k_scan<<<1, 256, 0, stream>>>(pong, h0, pend, sst);
    k_final<<<NC / 2, 256, 0, stream>>>(A, sst, U);
}